// PlayerMovementGNN_82308753260856
// MI455X (gfx1250) — compile-verified
//
#include <hip/hip_runtime.h>

typedef float v2f __attribute__((ext_vector_type(2)));
typedef float v8f __attribute__((ext_vector_type(8)));

#define THREADS 256

// ---------------- degree / normalization ----------------

__global__ void k_deg_init(float* __restrict__ deg, int n) {
    int i = blockIdx.x * blockDim.x + threadIdx.x;
    if (i < n) deg[i] = 1.0f;   // self-loop weight
}

__global__ void k_deg_accum(const int* __restrict__ col, const float* __restrict__ ew,
                            float* __restrict__ deg, int E) {
    int e = blockIdx.x * blockDim.x + threadIdx.x;
    if (e < E) atomicAdd(&deg[col[e]], ew[e]);
}

__global__ void k_dinv(float* __restrict__ deg, int n) {
    int i = blockIdx.x * blockDim.x + threadIdx.x;
    if (i < n) {
        float d = deg[i];
        deg[i] = (d > 0.0f) ? rsqrtf(d) : 0.0f;
    }
}

// ---------------- dense GEMM via f32 WMMA ----------------
// out[N x 64] = act(A[N x K]) @ W[K x 64]
// One wave computes one 16x16 tile of out, chaining V_WMMA_F32_16X16X4_F32 over K.
// A-frag (16x4, 32b): lanes 0-15 rows M=lane  VGPR0=K0 VGPR1=K1; lanes 16-31 K2/K3.
// B-frag (4x16, 32b): lanes 0-15 cols N=lane  VGPR0=row K0 VGPR1=row K1; lanes 16-31 rows K2/K3.
// C/D (16x16 f32): VGPR r -> lanes 0-15 M=r, lanes 16-31 M=r+8; N=lane&15.
template <int K, bool RELU_IN>
__global__ void k_gemm_wmma(const float* __restrict__ A, const float* __restrict__ W,
                            float* __restrict__ out, int nTiles) {
    const int lane = threadIdx.x & 31;
    const int wave = threadIdx.x >> 5;
    const int tile = blockIdx.x * (blockDim.x >> 5) + wave;
    if (tile >= nTiles) return;                 // wave-uniform exit: EXEC stays all-1s

    const int NT = 64 / 16;                     // column tiles
    const int m0 = (tile / NT) * 16;
    const int n0 = (tile % NT) * 16;
    const int mlane = lane & 15;
    const int hi = lane >> 4;                   // 0: K pair {0,1}; 1: K pair {2,3}

    v8f acc = {};
#pragma unroll
    for (int kk = 0; kk < K / 4; ++kk) {
        const int kbase = kk * 4 + hi * 2;
        // A fragment: row m0+mlane, columns kbase, kbase+1 (8B-aligned pair)
        const float* ap = A + (size_t)(m0 + mlane) * K + kbase;
        v2f a;
        a.x = ap[0];
        a.y = ap[1];
        if (RELU_IN) {
            a.x = fmaxf(a.x, 0.0f);
            a.y = fmaxf(a.y, 0.0f);
        }
        // B fragment: rows kbase, kbase+1; column n0+mlane
        const float* bp = W + (size_t)kbase * 64 + n0 + mlane;
        v2f b;
        b.x = bp[0];
        b.y = bp[64];
        acc = __builtin_amdgcn_wmma_f32_16x16x4_f32(
            /*neg_a=*/false, a, /*neg_b=*/false, b,
            /*c_mod=*/(short)0, acc, /*reuse_a=*/false, /*reuse_b=*/false);
    }
#pragma unroll
    for (int r = 0; r < 8; ++r) {
        const int row = m0 + r + hi * 8;
        out[(size_t)row * 64 + n0 + mlane] = acc[r];
    }
}

// ---------------- self-loop + bias init:  out = h * dinv^2 + bias ----------------
__global__ void k_agg_init(const float* __restrict__ h, const float* __restrict__ dinv,
                           const float* __restrict__ bias, float* __restrict__ out, int n) {
    int idx = blockIdx.x * blockDim.x + threadIdx.x;   // over n*16 float4 groups
    if (idx >= n * 16) return;
    int node = idx >> 4;
    int q = idx & 15;
    float d = dinv[node];
    float d2 = d * d;
    float4 hv = ((const float4*)h)[idx];
    float4 bv = ((const float4*)bias)[q];
    float4 o;
    o.x = fmaf(hv.x, d2, bv.x);
    o.y = fmaf(hv.y, d2, bv.y);
    o.z = fmaf(hv.z, d2, bv.z);
    o.w = fmaf(hv.w, d2, bv.w);
    ((float4*)out)[idx] = o;
}

// ---------------- edge scatter: 16 lanes per edge, 64 channels ----------------
__global__ void k_scatter(const int* __restrict__ row, const int* __restrict__ col,
                          const float* __restrict__ ew, const float* __restrict__ dinv,
                          const float* __restrict__ h, float* __restrict__ out, int E) {
    int t = blockIdx.x * blockDim.x + threadIdx.x;
    int e = t >> 4;
    int sub = t & 15;
    if (e >= E) return;
    int r = row[e];
    int c = col[e];
    float nw = dinv[r] * ew[e] * dinv[c];
    float4 hv = ((const float4*)(h + (size_t)r * 64))[sub];
    float* op = out + (size_t)c * 64 + sub * 4;
    atomicAdd(op + 0, hv.x * nw);
    atomicAdd(op + 1, hv.y * nw);
    atomicAdd(op + 2, hv.z * nw);
    atomicAdd(op + 3, hv.w * nw);
}

// ---------------- final linear: out[N x 2] = relu(h[N x 64]) @ Wl + bl ----------------
__global__ void k_final(const float* __restrict__ h, const float* __restrict__ Wl,
                        const float* __restrict__ bl, float* __restrict__ out, int n) {
    int node = blockIdx.x * blockDim.x + threadIdx.x;
    if (node >= n) return;
    float a0 = bl[0], a1 = bl[1];
    const float* hp = h + (size_t)node * 64;
#pragma unroll
    for (int k = 0; k < 64; ++k) {
        float v = fmaxf(hp[k], 0.0f);
        a0 = fmaf(v, Wl[2 * k + 0], a0);
        a1 = fmaf(v, Wl[2 * k + 1], a1);
    }
    out[(size_t)node * 2 + 0] = a0;
    out[(size_t)node * 2 + 1] = a1;
}

// ---------------- driver ----------------

extern "C" void kernel_launch(void* const* d_in, const int* in_sizes, int n_in,
                              void* d_out, int out_size, void* d_ws, size_t ws_size,
                              hipStream_t stream) {
    const float* x   = (const float*)d_in[0];   // [N,16]
    const int*   ei  = (const int*)d_in[1];     // [2,E]
    const float* ew  = (const float*)d_in[2];   // [E]
    const float* W1  = (const float*)d_in[3];   // [16,64]
    const float* b1  = (const float*)d_in[4];   // [64]
    const float* W2  = (const float*)d_in[5];   // [64,64]
    const float* b2  = (const float*)d_in[6];   // [64]
    const float* Wl  = (const float*)d_in[7];   // [64,2]
    const float* bl  = (const float*)d_in[8];   // [2]
    float* outp = (float*)d_out;

    const int N = in_sizes[0] / 16;
    const int E = in_sizes[2];
    const int* row = ei;
    const int* col = ei + E;

    // workspace: dinv [N] | B0 [N*64] | B1 [N*64]
    float* dinv = (float*)d_ws;
    float* B0 = dinv + N;
    float* B1 = B0 + (size_t)N * 64;

    const int nTiles = (N / 16) * (64 / 16);
    const int gemmBlocks = (nTiles + (THREADS / 32) - 1) / (THREADS / 32);

    // normalization
    k_deg_init<<<(N + THREADS - 1) / THREADS, THREADS, 0, stream>>>(dinv, N);
    k_deg_accum<<<(E + THREADS - 1) / THREADS, THREADS, 0, stream>>>(col, ew, dinv, E);
    k_dinv<<<(N + THREADS - 1) / THREADS, THREADS, 0, stream>>>(dinv, N);

    // layer 1: B0 = x @ W1 ; B1 = B0*dinv^2 + b1 ; scatter edges B0 -> B1
    k_gemm_wmma<16, false><<<gemmBlocks, THREADS, 0, stream>>>(x, W1, B0, nTiles);
    k_agg_init<<<(N * 16 + THREADS - 1) / THREADS, THREADS, 0, stream>>>(B0, dinv, b1, B1, N);
    k_scatter<<<((size_t)E * 16 + THREADS - 1) / THREADS, THREADS, 0, stream>>>(
        row, col, ew, dinv, B0, B1, E);

    // layer 2: B0 = relu(B1) @ W2 ; B1 = B0*dinv^2 + b2 ; scatter edges B0 -> B1
    k_gemm_wmma<64, true><<<gemmBlocks, THREADS, 0, stream>>>(B1, W2, B0, nTiles);
    k_agg_init<<<(N * 16 + THREADS - 1) / THREADS, THREADS, 0, stream>>>(B0, dinv, b2, B1, N);
    k_scatter<<<((size_t)E * 16 + THREADS - 1) / THREADS, THREADS, 0, stream>>>(
        row, col, ew, dinv, B0, B1, E);

    // final: out = relu(B1) @ Wl + bl
    k_final<<<(N + THREADS - 1) / THREADS, THREADS, 0, stream>>>(B1, Wl, bl, outp, N);
}